// StationR2Loss_54546084659704
// MI455X (gfx1250) — compile-verified
//
#include <hip/hip_runtime.h>

#define NSTATIONS 4096
#define EPSF 1e-8f
#define ACCUM_BLOCKS 256
#define BLOCK_THREADS 256

typedef __attribute__((ext_vector_type(2))) float v2f;
typedef __attribute__((ext_vector_type(8))) float v8f;

// ---------------------------------------------------------------------------
// Kernel 0: zero the global accumulators in workspace (ws is poisoned 0xAA).
// ---------------------------------------------------------------------------
__global__ void StationR2_zero_ws(float* __restrict__ ws, int n) {
    int i = blockIdx.x * blockDim.x + threadIdx.x;
    if (i < n) ws[i] = 0.0f;
}

// ---------------------------------------------------------------------------
// Kernel 1: LDS-privatized segmented accumulation.
//   ws layout (planar): [0..S)   = count
//                       [S..2S)  = sum(t)
//                       [2S..3S) = sum(t^2)
//                       [3S..4S) = sum((t-p)^2)
// LDS layout (interleaved): s_acc[sid*4 + {0,1,2,3}] -> 4 consecutive banks.
// ---------------------------------------------------------------------------
__global__ __launch_bounds__(BLOCK_THREADS) void StationR2_accum(
    const float* __restrict__ pred,
    const float* __restrict__ targ,
    const int*  __restrict__ sid,
    float* __restrict__ ws,
    int n)
{
    __shared__ float s_acc[NSTATIONS * 4];   // 64 KB

    for (int i = threadIdx.x; i < NSTATIONS * 4; i += BLOCK_THREADS)
        s_acc[i] = 0.0f;
    __syncthreads();

    const int tid    = blockIdx.x * BLOCK_THREADS + threadIdx.x;
    const int stride = gridDim.x * BLOCK_THREADS;

    const float4* p4 = (const float4*)pred;
    const float4* t4 = (const float4*)targ;
    const int4*   i4 = (const int4*)sid;
    const int n4 = n >> 2;

    for (int i = tid; i < n4; i += stride) {
        int nx = i + stride;
        if (nx < n4) {                        // gfx1250: global_prefetch_b8
            __builtin_prefetch(&p4[nx], 0, 0);
            __builtin_prefetch(&t4[nx], 0, 0);
            __builtin_prefetch(&i4[nx], 0, 0);
        }
        float4 p = p4[i];
        float4 t = t4[i];
        int4   s = i4[i];

        {
            float d = t.x - p.x; int b = s.x << 2;
            atomicAdd(&s_acc[b + 0], 1.0f);
            atomicAdd(&s_acc[b + 1], t.x);
            atomicAdd(&s_acc[b + 2], t.x * t.x);
            atomicAdd(&s_acc[b + 3], d * d);
        }
        {
            float d = t.y - p.y; int b = s.y << 2;
            atomicAdd(&s_acc[b + 0], 1.0f);
            atomicAdd(&s_acc[b + 1], t.y);
            atomicAdd(&s_acc[b + 2], t.y * t.y);
            atomicAdd(&s_acc[b + 3], d * d);
        }
        {
            float d = t.z - p.z; int b = s.z << 2;
            atomicAdd(&s_acc[b + 0], 1.0f);
            atomicAdd(&s_acc[b + 1], t.z);
            atomicAdd(&s_acc[b + 2], t.z * t.z);
            atomicAdd(&s_acc[b + 3], d * d);
        }
        {
            float d = t.w - p.w; int b = s.w << 2;
            atomicAdd(&s_acc[b + 0], 1.0f);
            atomicAdd(&s_acc[b + 1], t.w);
            atomicAdd(&s_acc[b + 2], t.w * t.w);
            atomicAdd(&s_acc[b + 3], d * d);
        }
    }

    // Scalar tail (n may not be a multiple of 4).
    for (int i = (n4 << 2) + tid; i < n; i += stride) {
        float p = pred[i], t = targ[i];
        float d = t - p;
        int b = sid[i] << 2;
        atomicAdd(&s_acc[b + 0], 1.0f);
        atomicAdd(&s_acc[b + 1], t);
        atomicAdd(&s_acc[b + 2], t * t);
        atomicAdd(&s_acc[b + 3], d * d);
    }

    __syncthreads();

    // Flush block-private partials to global accumulators (HW f32 atomics).
    for (int s = threadIdx.x; s < NSTATIONS; s += BLOCK_THREADS) {
        float c = s_acc[s * 4 + 0];
        if (c != 0.0f) {
            unsafeAtomicAdd(&ws[s],                 c);
            unsafeAtomicAdd(&ws[NSTATIONS + s],     s_acc[s * 4 + 1]);
            unsafeAtomicAdd(&ws[2 * NSTATIONS + s], s_acc[s * 4 + 2]);
            unsafeAtomicAdd(&ws[3 * NSTATIONS + s], s_acc[s * 4 + 3]);
        }
    }
}

// ---------------------------------------------------------------------------
// Kernel 2: per-station finalize + WMMA-assisted block reduction.
// Reduction trick: D = A x B + 0 with B = ones(4x16). Row-sums of A land in
// every column of D. A (16x4 f32) holds 64 partials: lane L supplies
// {A[L%16][K],A[L%16][K+1]} (K=0 for lanes 0-15, K=2 for lanes 16-31).
// Total = sum_i D[i][0] = (sum of 8 C/D VGPRs on lane 0) + (same on lane 16).
// ---------------------------------------------------------------------------
__global__ __launch_bounds__(BLOCK_THREADS) void StationR2_finalize(
    const float* __restrict__ ws,
    float* __restrict__ out)
{
    __shared__ float red_l[BLOCK_THREADS];
    __shared__ float red_p[BLOCK_THREADS];

    float lsum = 0.0f, psum = 0.0f;
    for (int s = threadIdx.x; s < NSTATIONS; s += BLOCK_THREADS) {
        float c    = ws[s];
        float st   = ws[NSTATIONS + s];
        float st2  = ws[2 * NSTATIONS + s];
        float ssr  = ws[3 * NSTATIONS + s];

        float safe   = fmaxf(c, 1.0f);
        float mse    = ssr / safe;
        float ss_tot = st2 - (st * st) / safe;
        bool  ok     = ss_tot > EPSF;
        float r2     = 1.0f - ssr / (ok ? ss_tot : 1.0f);
        float r2l    = 1.0f - fminf(fmaxf(r2, -1.0f), 1.0f);
        float big    = ok ? r2l : mse;
        float per    = (c >= 5.0f) ? big : mse;
        bool  present = c > 0.0f;
        lsum += present ? per : 0.0f;
        psum += present ? 1.0f : 0.0f;
    }
    red_l[threadIdx.x] = lsum;
    red_p[threadIdx.x] = psum;
    __syncthreads();

    if (threadIdx.x < 32) {                 // wave 0 only; EXEC = all 32 lanes
        const int lane = threadIdx.x;
        const int base = lane * 8;
        float x0 = 0.0f, x1 = 0.0f, y0 = 0.0f, y1 = 0.0f;
#pragma unroll
        for (int k = 0; k < 4; ++k) { x0 += red_l[base + k];     y0 += red_p[base + k]; }
#pragma unroll
        for (int k = 4; k < 8; ++k) { x1 += red_l[base + k];     y1 += red_p[base + k]; }

        v2f aL = {x0, x1};
        v2f aP = {y0, y1};
        v2f bOnes = {1.0f, 1.0f};
        v8f c0 = {};

        v8f dL = __builtin_amdgcn_wmma_f32_16x16x4_f32(
            false, aL, false, bOnes, (short)0, c0, false, false);
        v8f dP = __builtin_amdgcn_wmma_f32_16x16x4_f32(
            false, aP, false, bOnes, (short)0, c0, false, false);

        float sL = dL[0] + dL[1] + dL[2] + dL[3] + dL[4] + dL[5] + dL[6] + dL[7];
        float sP = dP[0] + dP[1] + dP[2] + dP[3] + dP[4] + dP[5] + dP[6] + dP[7];

        float totL = __shfl(sL, 0, 32) + __shfl(sL, 16, 32);
        float totP = __shfl(sP, 0, 32) + __shfl(sP, 16, 32);

        if (lane == 0)
            out[0] = totL / fmaxf(totP, 1.0f);
    }
}

// ---------------------------------------------------------------------------
extern "C" void kernel_launch(void* const* d_in, const int* in_sizes, int n_in,
                              void* d_out, int out_size, void* d_ws, size_t ws_size,
                              hipStream_t stream)
{
    const float* pred = (const float*)d_in[0];
    const float* targ = (const float*)d_in[1];
    const int*   sid  = (const int*)d_in[2];
    float* ws  = (float*)d_ws;
    float* out = (float*)d_out;
    const int n = in_sizes[0];

    StationR2_zero_ws<<<(4 * NSTATIONS + 255) / 256, 256, 0, stream>>>(ws, 4 * NSTATIONS);
    StationR2_accum<<<ACCUM_BLOCKS, BLOCK_THREADS, 0, stream>>>(pred, targ, sid, ws, n);
    StationR2_finalize<<<1, BLOCK_THREADS, 0, stream>>>(ws, out);
}